// Kernel_2066129463177083751_28716151341299
// MI455X (gfx1250) — compile-verified
//
#include <hip/hip_runtime.h>

#define NN 32
#define CC 256
#define HH 64
#define WWD 64
#define HWSZ 4096   // H*W
#define NCOL 2048   // N*W gemm columns
#define MROWS 1600  // 25*H

typedef __attribute__((ext_vector_type(2))) float v2f;
typedef __attribute__((ext_vector_type(8))) float v8f;

__device__ __forceinline__ void atomicMaxF(float* addr, float val) {
  if (val >= 0.0f) atomicMax((int*)addr, __float_as_int(val));
  else             atomicMin((unsigned int*)addr, __float_as_uint(val));
}

// w2bar[c] = mean_o W2[o,c]   (folds the t2 GEMM through the channel mean)
__global__ void k_w2bar(const float* __restrict__ W2, float* __restrict__ w2bar) {
  int c = threadIdx.x;
  float s = 0.f;
  for (int o = 0; o < CC; ++o) s += W2[o * CC + c];
  w2bar[c] = s * (1.0f / 256.0f);
}

// t3[n,c] = max(0, max_{h,w} x[n,c,h,w])
__global__ void k_t3(const float* __restrict__ x, float* __restrict__ t3) {
  int b = blockIdx.x;  // n*C + c
  const float* base = x + (size_t)b * HWSZ;
  int tid = threadIdx.x;
  float m = 0.f;
  for (int i = tid; i < HWSZ; i += 256) m = fmaxf(m, base[i]);
  for (int off = 16; off > 0; off >>= 1) m = fmaxf(m, __shfl_xor(m, off));
  __shared__ float red[8];
  if ((tid & 31) == 0) red[tid >> 5] = m;
  __syncthreads();
  if (tid == 0) {
    float mm = red[0];
    for (int i = 1; i < 8; ++i) mm = fmaxf(mm, red[i]);
    t3[b] = mm;
  }
}

// t5[n,hw] = p4[hw] * sum_c w2bar[c]*relu(x[n,c,hw])
__global__ void k_t5(const float* __restrict__ x, const float* __restrict__ w2bar,
                     const float* __restrict__ p4, float* __restrict__ t5) {
  int n = blockIdx.y;
  int hw = blockIdx.x * 256 + threadIdx.x;
  const float* base = x + (size_t)n * CC * HWSZ + hw;
  float s = 0.f;
  for (int c = 0; c < CC; ++c) s = fmaf(w2bar[c], fmaxf(base[c * HWSZ], 0.f), s);
  t5[n * HWSZ + hw] = p4[hw] * s;
}

// t7[n,k,w] = max_h of shifted t5 (dilated 5x5 unfold, then max over h)
__global__ void k_t7(const float* __restrict__ t5, float* __restrict__ t7) {
  int gid = blockIdx.x * 256 + threadIdx.x;
  if (gid >= NN * 25 * WWD) return;
  int w = gid & 63;
  int k = (gid >> 6) % 25;
  int n = gid / (25 * 64);
  int ki = k / 5, kj = k % 5;
  int ww = w + 3 * kj - 6;
  float acc;
  if (ww < 0 || ww >= WWD) acc = 0.f;
  else {
    acc = -__builtin_inff();
    const float* tn = t5 + n * HWSZ;
    for (int h = 0; h < HH; ++h) {
      int hh = h + 3 * ki - 6;
      float v = (hh >= 0 && hh < HH) ? tn[hh * WWD + ww] : 0.f;
      acc = fmaxf(acc, v);
    }
  }
  t7[gid] = acc;  // gid == n*1600 + k*64 + w
}

// t8B[c, col] = sum_k W8[c,k]*t7[n,k,w], col = n*64+w  (B matrix for the GEMM)
__global__ void k_t8(const float* __restrict__ W8, const float* __restrict__ t7,
                     float* __restrict__ t8B) {
  int gid = blockIdx.x * 256 + threadIdx.x;  // c*2048 + col
  int col = gid & (NCOL - 1);
  int c = gid >> 11;
  int n = col >> 6, w = col & 63;
  float s = 0.f;
  const float* wr = W8 + c * 25;
  const float* tn = t7 + n * (25 * WWD) + w;
  for (int k = 0; k < 25; ++k) s = fmaf(wr[k], tn[k * WWD], s);
  t8B[gid] = s;
}

__global__ void k_initmx(float* __restrict__ mx) {
  int gid = blockIdx.x * 256 + threadIdx.x;
  if (gid < NCOL) mx[gid] = -__builtin_inff();
}

// D = W10(1600x256) x t8B(256x2048) via v_wmma_f32_16x16x4_f32,
// fused epilogue: gelu(exact erf) + t6(shifted t5) add + column max -> mx
__global__ __launch_bounds__(32) void k_gemm_t10(const float* __restrict__ W10,
                                                 const float* __restrict__ t8B,
                                                 const float* __restrict__ t5,
                                                 float* __restrict__ mx) {
  const int Mt = blockIdx.x;   // 0..99
  const int Nt = blockIdx.y;   // 0..127
  const int lane = threadIdx.x;
  const int row = lane & 15;
  const int hi = lane >> 4;
  const int m0 = Mt * 16;
  const int col0 = Nt * 16;
  // A lane layout: lane holds A[m0+row, k0+2*hi+{0,1}]
  const float* Ap = W10 + (m0 + row) * CC + 2 * hi;
  // B lane layout: lane holds B[k0+2*hi+{0,1}, col0+row]
  const float* Bp = t8B + (2 * hi) * NCOL + col0 + row;
  v8f acc = {};
  for (int k0 = 0; k0 < CC; k0 += 4) {
    v2f a = *(const v2f*)(Ap + k0);
    v2f b;
    b.x = Bp[k0 * NCOL];
    b.y = Bp[(k0 + 1) * NCOL];
    acc = __builtin_amdgcn_wmma_f32_16x16x4_f32(false, a, false, b, (short)0, acc,
                                                false, false);
  }
  // epilogue: per-element gelu + t6flat add, then column max into mx
  const int col = col0 + row;
  const int n = col >> 6, w = col & 63;
  const int kk = m0 >> 6;  // row tile lies inside one 64-block -> k constant
  const int ki = kk / 5, kj = kk % 5;
  const int hbase = (m0 & 63) + 8 * hi;
  const int ww = w + 3 * kj - 6;
  const float* tn = t5 + n * HWSZ;
  float vmax = -__builtin_inff();
#pragma unroll
  for (int v = 0; v < 8; ++v) {
    float g = acc[v];
    g = 0.5f * g * (1.0f + erff(g * 0.70710678118654752f));  // exact-erf gelu
    int hh = hbase + v + 3 * ki - 6;
    float t6v = (hh >= 0 && hh < HH && ww >= 0 && ww < WWD) ? tn[hh * WWD + ww] : 0.f;
    vmax = fmaxf(vmax, g + t6v);
  }
  vmax = fmaxf(vmax, __shfl_xor(vmax, 16));  // combine row halves (same column)
  if (lane < 16) atomicMaxF(&mx[col], vmax);
}

// t17[n,w] = (1/7) * sum_j exp(mx[n, w+3j-9]) over valid shifts (exp∘max == max∘exp)
__global__ void k_t17(const float* __restrict__ mx, float* __restrict__ t17) {
  int gid = blockIdx.x * 256 + threadIdx.x;
  if (gid >= NCOL) return;
  int n = gid >> 6, w = gid & 63;
  float s = 0.f;
  for (int j = 0; j < 7; ++j) {
    int ww = w + 3 * j - 9;
    if (ww >= 0 && ww < WWD) s += expf(mx[n * WWD + ww]);
  }
  t17[gid] = s * (1.0f / 7.0f);
}

// t13[n,hw] = p13[hw] * sum_c t3[n,c]*x[n,c,hw]
__global__ void k_t13(const float* __restrict__ x, const float* __restrict__ t3,
                      const float* __restrict__ p13, float* __restrict__ t13) {
  int n = blockIdx.y;
  int hw = blockIdx.x * 256 + threadIdx.x;
  const float* base = x + (size_t)n * CC * HWSZ + hw;
  const float* tn = t3 + n * CC;
  float s = 0.f;
  for (int c = 0; c < CC; ++c) s = fmaf(tn[c], base[c * HWSZ], s);
  t13[n * HWSZ + hw] = p13[hw] * s;
}

// out[n,c,h,w] = t13[n,h,w] - W18[c*64+h]*t17[n,w]   (float4 along w)
__global__ void k_out(const float* __restrict__ t13, const float* __restrict__ W18,
                      const float* __restrict__ t17, float* __restrict__ out) {
  long long t = (long long)blockIdx.x * 256 + threadIdx.x;  // [0, 8388608)
  int w4 = (int)(t & 15) * 4;
  int h = (int)(t >> 4) & 63;
  int c = (int)(t >> 10) & 255;
  int n = (int)(t >> 18);
  const float4 a = *(const float4*)(t13 + n * HWSZ + h * WWD + w4);
  const float4 b = *(const float4*)(t17 + n * WWD + w4);
  float wv = W18[c * HH + h];
  float4 r;
  r.x = a.x - wv * b.x;
  r.y = a.y - wv * b.y;
  r.z = a.z - wv * b.z;
  r.w = a.w - wv * b.w;
  *(float4*)(out + t * 4) = r;
}

extern "C" void kernel_launch(void* const* d_in, const int* in_sizes, int n_in,
                              void* d_out, int out_size, void* d_ws, size_t ws_size,
                              hipStream_t stream) {
  (void)in_sizes; (void)n_in; (void)out_size; (void)ws_size;
  const float* x   = (const float*)d_in[0];
  const float* W2  = (const float*)d_in[1];
  const float* W8  = (const float*)d_in[2];
  const float* W10 = (const float*)d_in[3];
  const float* W18 = (const float*)d_in[4];
  const float* p4  = (const float*)d_in[5];
  const float* p13 = (const float*)d_in[6];
  float* out = (float*)d_out;

  float* ws    = (float*)d_ws;
  float* w2bar = ws;  ws += 256;
  float* t5    = ws;  ws += NN * HWSZ;
  float* t3    = ws;  ws += NN * CC;
  float* t7    = ws;  ws += NN * 25 * WWD;
  float* t8B   = ws;  ws += CC * NCOL;
  float* mx    = ws;  ws += NCOL;
  float* t17   = ws;  ws += NCOL;
  float* t13   = ws;  ws += NN * HWSZ;

  hipLaunchKernelGGL(k_w2bar, dim3(1), dim3(256), 0, stream, W2, w2bar);
  hipLaunchKernelGGL(k_t3, dim3(NN * CC), dim3(256), 0, stream, x, t3);
  hipLaunchKernelGGL(k_t5, dim3(16, NN), dim3(256), 0, stream, x, w2bar, p4, t5);
  hipLaunchKernelGGL(k_t7, dim3((NN * 25 * WWD + 255) / 256), dim3(256), 0, stream, t5, t7);
  hipLaunchKernelGGL(k_t8, dim3(CC * NCOL / 256), dim3(256), 0, stream, W8, t7, t8B);
  hipLaunchKernelGGL(k_initmx, dim3(8), dim3(256), 0, stream, mx);
  hipLaunchKernelGGL(k_gemm_t10, dim3(100, 128), dim3(32), 0, stream, W10, t8B, t5, mx);
  hipLaunchKernelGGL(k_t17, dim3(8), dim3(256), 0, stream, mx, t17);
  hipLaunchKernelGGL(k_t13, dim3(16, NN), dim3(256), 0, stream, x, t3, p13, t13);
  hipLaunchKernelGGL(k_out, dim3(32768), dim3(256), 0, stream, t13, W18, t17, out);
}